// PolicyMap_49770081026298
// MI455X (gfx1250) — compile-verified
//
#include <hip/hip_runtime.h>
#include <stdint.h>

// PolicyMap: out[b, j] = x_flat[b, rows[j]] where rows[j] = argmax_i W[i, j]
// (W is a fixed one-hot selection matrix). Zero-FLOP gather at HBM roofline
// (~142 MB total traffic ~ 6 us @ 23.3 TB/s) instead of a 71-GFLOP f32 GEMM.
// CDNA5-specific path: gfx1250 async global->LDS b128 copies (ASYNCcnt) for
// row staging, drained with s_wait_asynccnt.

#define N_PLANES 73
#define FLAT     4672      // 73 * 8 * 8
#define N_MOVES  1858
#define BATCH    4096

// ---------------------------------------------------------------------------
// Kernel (placed FIRST so the disasm snippet shows it): one block per batch
// row. Stage the 4672-float row into LDS with gfx1250 async global->LDS b128
// copies (no VGPR round-trip), drain with s_wait_asynccnt, barrier, then
// gather out[b,j] = lds[rows[j]] with coalesced float2 stores.
// LDS use: 18688 B per block (WGP has 320 KB -> high occupancy).
// ---------------------------------------------------------------------------
__global__ __launch_bounds__(256) void gather_policy(const float* __restrict__ x,
                                                     const int* __restrict__ rows,
                                                     float* __restrict__ out) {
    __shared__ float tile[FLAT];

    const uint32_t b   = blockIdx.x;
    const uint32_t tid = threadIdx.x;

    // Low 32 bits of a generic LDS pointer == LDS byte offset on amdgcn.
    const uint32_t lds_base = (uint32_t)(uintptr_t)(&tile[0]);
    const uint64_t grow     = (uint64_t)(uintptr_t)(x + (size_t)b * FLAT);

    // 4672 floats = 1168 x b128 transfers; row base is 16B-aligned
    // (b * 18688 bytes). Each lane issues its own async copy.
#pragma unroll 2
    for (uint32_t k = tid; k < (uint32_t)(FLAT / 4); k += 256u) {
        const uint32_t lds_addr = lds_base + k * 16u;
        const uint64_t gaddr    = grow + (uint64_t)k * 16u;
        asm volatile("global_load_async_to_lds_b128 %0, %1, off"
                     :: "v"(lds_addr), "v"(gaddr)
                     : "memory");
    }
    // Drain this wave's async copies, then sync waves so all of LDS is valid.
    asm volatile("s_wait_asynccnt 0" ::: "memory");
    __syncthreads();

    // rows[] is 7.4 KB, reused by all 4096 blocks -> lives in L2.
    // 1858 = 2 * 929: emit paired gathers + one float2 store per step.
    // Row byte offset b*7432 is 8B-aligned, so float2 stores are legal.
    float* __restrict__ orow = out + (size_t)b * N_MOVES;
#pragma unroll 2
    for (uint32_t p = tid; p < (uint32_t)(N_MOVES / 2); p += 256u) {
        const uint32_t j0 = 2u * p;
        float2 v;
        v.x = tile[rows[j0]];
        v.y = tile[rows[j0 + 1u]];
        *reinterpret_cast<float2*>(orow + j0) = v;
    }
}

// ---------------------------------------------------------------------------
// Decode the one-hot matrix W[FLAT, N_MOVES] (row-major) into rows[N_MOVES].
// Exactly one nonzero per column -> exactly one writer per rows[j], no
// atomics needed. Fully-coalesced float4 streaming of 34.7 MB.
// ---------------------------------------------------------------------------
__global__ __launch_bounds__(256) void extract_rows(const float* __restrict__ W,
                                                    int* __restrict__ rows) {
    const uint32_t total = (uint32_t)FLAT * (uint32_t)N_MOVES;   // divisible by 4
    const uint32_t base  = (blockIdx.x * 256u + threadIdx.x) * 4u;
    if (base >= total) return;

    const float4 v = *reinterpret_cast<const float4*>(W + base);
    const float vals[4] = {v.x, v.y, v.z, v.w};
#pragma unroll
    for (int k = 0; k < 4; ++k) {
        if (vals[k] != 0.0f) {
            const uint32_t idx = base + (uint32_t)k;
            const uint32_t i   = idx / (uint32_t)N_MOVES;        // magic-mul div
            const uint32_t j   = idx - i * (uint32_t)N_MOVES;
            rows[j] = (int)i;
        }
    }
}

// ---------------------------------------------------------------------------
extern "C" void kernel_launch(void* const* d_in, const int* in_sizes, int n_in,
                              void* d_out, int out_size, void* d_ws, size_t ws_size,
                              hipStream_t stream) {
    const float* x = (const float*)d_in[0];   // [4096, 73, 8, 8] f32
    const float* W = (const float*)d_in[1];   // [4672, 1858] f32 one-hot
    float*       o = (float*)d_out;           // [4096, 1858] f32
    int*      rows = (int*)d_ws;              // 1858 ints of scratch

    const uint32_t total_vec4 = ((uint32_t)FLAT * (uint32_t)N_MOVES) / 4u;
    const uint32_t g1 = (total_vec4 + 255u) / 256u;

    extract_rows<<<dim3(g1), dim3(256), 0, stream>>>(W, rows);
    gather_policy<<<dim3(BATCH), dim3(256), 0, stream>>>(x, rows, o);
}